// Find_1915555414390
// MI455X (gfx1250) — compile-verified
//
#include <hip/hip_runtime.h>

typedef __attribute__((ext_vector_type(2))) float v2f;
typedef __attribute__((ext_vector_type(8))) float v8f;

#define NB   512   // batches
#define NC   512   // channels (reduction dim)
#define HW   196   // 14*14 pixels
#define NTIL 13    // ceil(196/16) pixel tiles of 16

__global__ __launch_bounds__(256) void masks_gemv_wmma(
    const float* __restrict__ features,   // [B, C, H, W] fp32
    const int*   __restrict__ cidx,       // [B]
    const float* __restrict__ weight,     // [K, C] fp32 (1x1 conv)
    float*       __restrict__ out)        // [B, 1, H, W] fp32
{
    __shared__ __align__(16) float wsm[NC];   // gathered weight row, 2 KB

    const int b = blockIdx.x;
    const int t = threadIdx.x;

    // ---- gather weight[c[b], :] into LDS (256 threads x float2 = 512 floats)
    const float2* wk2 = (const float2*)(weight + (size_t)cidx[b] * NC);
    ((float2*)wsm)[t] = wk2[t];
    __syncthreads();

    const int lane    = t & 31;
    const int wave    = t >> 5;            // wave32: 8 waves per block
    const int halfsel = (lane >> 4) << 1;  // 0 for lanes 0-15 (K=0,1), 2 for lanes 16-31 (K=2,3)
    const int col     = lane & 15;         // N index within 16-pixel tile

    const float* F = features + (size_t)b * NC * HW;

    for (int tile = wave; tile < NTIL; tile += 8) {
        const int p0  = tile * 16;
        const int pix = p0 + col;
        const int pc  = (pix < HW) ? pix : (HW - 1);   // clamp: keep EXEC all-1s for WMMA

        v8f acc = {0.f, 0.f, 0.f, 0.f, 0.f, 0.f, 0.f, 0.f};

#pragma unroll 4
        for (int k0 = 0; k0 < NC; k0 += 4) {
            // A: 16x4 f32, all rows = w[k0..k0+3] (broadcast over M).
            // Lane layout (ISA 7.12.2): lanes 0-15 hold K=0,1; lanes 16-31 hold K=2,3.
            v2f a = *(const v2f*)&wsm[k0 + halfsel];

            // B: 4x16 f32 tile f[k0..k0+3, p0..p0+15].
            // VGPR0 = K = halfsel, VGPR1 = K = halfsel+1, N striped over lanes.
            const float* fb = F + (size_t)(k0 + halfsel) * HW + pc;
            v2f bv;
            bv.x = __builtin_nontemporal_load(fb);        // streamed once: NT hint
            bv.y = __builtin_nontemporal_load(fb + HW);

            acc = __builtin_amdgcn_wmma_f32_16x16x4_f32(
                      /*neg_a=*/false, a,
                      /*neg_b=*/false, bv,
                      /*c_mod=*/(short)0, acc,
                      /*reuse_a=*/false, /*reuse_b=*/false);
        }

        // D rows are all identical (A rows were broadcast). Row M=0 sits in
        // VGPR0 of lanes 0-15 (C/D layout: VGPR r, lanes 0-15 -> M=r).
        if (lane < 16 && pix < HW) {
            const float m = acc[0];
            const float s = 1.0f / (1.0f + __expf(-m));
            __builtin_nontemporal_store(s, out + (size_t)b * HW + pix);
        }
    }
}

extern "C" void kernel_launch(void* const* d_in, const int* in_sizes, int n_in,
                              void* d_out, int out_size, void* d_ws, size_t ws_size,
                              hipStream_t stream) {
    const float* features = (const float*)d_in[0];
    const int*   cidx     = (const int*)d_in[1];
    const float* weight   = (const float*)d_in[2];
    float*       out      = (float*)d_out;
    (void)in_sizes; (void)n_in; (void)out_size; (void)d_ws; (void)ws_size;

    masks_gemv_wmma<<<NB, 256, 0, stream>>>(features, cidx, weight, out);
}